// Encoder_GCNTCNTCN2_44573170598236
// MI455X (gfx1250) — compile-verified
//
#include <hip/hip_runtime.h>
#include <math.h>

typedef __attribute__((ext_vector_type(2))) float v2f;
typedef __attribute__((ext_vector_type(8))) float v8f;

#define WMMA_F32(a, b, c) \
  __builtin_amdgcn_wmma_f32_16x16x4_f32(false, (a), false, (b), (short)0, (c), false, false)

// ---------------------------------------------------------------------------
// Model constants
// ---------------------------------------------------------------------------
#define BS 64
#define JNT 24
#define FD 6
#define TT 60
#define VV 25
#define TV 1500        // T*V
#define HG 256         // gating hidden
#define EE 4           // experts
#define NSPLIT 75      // split-K: all K in {9000,96000,384000} divisible by 75*8

// ---------------------------------------------------------------------------
// 1. Build h0:(64,6,60,25) from x:(64,24,6,60) + label embedding
// ---------------------------------------------------------------------------
__global__ void k_build_input(const float* __restrict__ x, const int* __restrict__ y,
                              const float* __restrict__ lab, float* __restrict__ h) {
  int i = blockIdx.x * blockDim.x + threadIdx.x;
  if (i >= BS * FD * TT * VV) return;
  int v = i % VV;
  int t = (i / VV) % TT;
  int f = (i / (VV * TT)) % FD;
  int n = i / (FD * TT * VV);
  float val;
  if (v < JNT) val = x[(((size_t)n * JNT + v) * FD + f) * TT + t];
  else         val = lab[y[n] * FD + f];
  h[i] = val;
}

// ---------------------------------------------------------------------------
// 2. Gating GEMM #1, split-K: partial[s] = A(64 x chunk) @ B(chunk x 256)
//    chunk = K/NSPLIT is an exact multiple of 8 -> guard-free inner loop.
//    One wave per (Ntile, split); holds all 4 Mtile accumulators so each
//    g0w element is loaded exactly once from HBM.
//    B (g0w, up to 393MB, zero reuse) is loaded NON-TEMPORAL so the 64xK
//    activation matrix (<=98MB, 16x reuse across Ntiles) stays hot in the
//    192MB L2. Partials are write-once/read-once -> NT as well.
// ---------------------------------------------------------------------------
__global__ __launch_bounds__(32)
void k_gate0_splitk(const float* __restrict__ A, const float* __restrict__ B,
                    float* __restrict__ part, int K, int chunk) {
  int nt = blockIdx.x;            // 0..15  (N = 256)
  int s  = blockIdx.y;            // split
  int lane = threadIdx.x & 31;
  int l16 = lane & 15, hi = lane >> 4;
  int k0 = s * chunk;
  int k1 = k0 + chunk;
  int nb = nt * 16;
  const float* Bp = B + nb + l16;
  const float* A0 = A + (size_t)(0 * 16 + l16) * K;
  const float* A1 = A + (size_t)(1 * 16 + l16) * K;
  const float* A2 = A + (size_t)(2 * 16 + l16) * K;
  const float* A3 = A + (size_t)(3 * 16 + l16) * K;
  v8f c0 = {}, c1 = {}, c2 = {}, c3 = {};
  for (int k = k0; k < k1; k += 8) {
    int ka = k + 2 * hi;
    int kb = ka + 4;
    v2f bfa, bfb;
    bfa.x = __builtin_nontemporal_load(Bp + (size_t)ka * HG);
    bfa.y = __builtin_nontemporal_load(Bp + (size_t)(ka + 1) * HG);
    bfb.x = __builtin_nontemporal_load(Bp + (size_t)kb * HG);
    bfb.y = __builtin_nontemporal_load(Bp + (size_t)(kb + 1) * HG);
    v2f a0a = *(const v2f*)(A0 + ka), a0b = *(const v2f*)(A0 + kb);
    v2f a1a = *(const v2f*)(A1 + ka), a1b = *(const v2f*)(A1 + kb);
    v2f a2a = *(const v2f*)(A2 + ka), a2b = *(const v2f*)(A2 + kb);
    v2f a3a = *(const v2f*)(A3 + ka), a3b = *(const v2f*)(A3 + kb);
    c0 = WMMA_F32(a0a, bfa, c0);
    c1 = WMMA_F32(a1a, bfa, c1);
    c2 = WMMA_F32(a2a, bfa, c2);
    c3 = WMMA_F32(a3a, bfa, c3);
    c0 = WMMA_F32(a0b, bfb, c0);
    c1 = WMMA_F32(a1b, bfb, c1);
    c2 = WMMA_F32(a2b, bfb, c2);
    c3 = WMMA_F32(a3b, bfb, c3);
  }
  float* P = part + (size_t)s * BS * HG + nb + l16;
  int rbase = 8 * hi;
#pragma unroll
  for (int r = 0; r < 8; ++r) {
    __builtin_nontemporal_store(c0[r], P + (size_t)(0  + r + rbase) * HG);
    __builtin_nontemporal_store(c1[r], P + (size_t)(16 + r + rbase) * HG);
    __builtin_nontemporal_store(c2[r], P + (size_t)(32 + r + rbase) * HG);
    __builtin_nontemporal_store(c3[r], P + (size_t)(48 + r + rbase) * HG);
  }
}

// reduce partials + bias + ELU -> h1:(64,256)
__global__ void k_reduce_elu(const float* __restrict__ part, const float* __restrict__ bias,
                             float* __restrict__ out, int S) {
  int i = blockIdx.x * blockDim.x + threadIdx.x;
  if (i >= BS * HG) return;
  float s = 0.f;
  for (int j = 0; j < S; ++j) s += __builtin_nontemporal_load(part + (size_t)j * BS * HG + i);
  s += bias[i & (HG - 1)];
  out[i] = (s > 0.f) ? s : expm1f(s);
}

// ---------------------------------------------------------------------------
// 3. Generic 64-row WMMA GEMM: C = act(A(64xK) @ B(KxN) + bias), act: 0=none,1=ELU
//    Guard-free main loop; single guarded tail step (only for K=150).
// ---------------------------------------------------------------------------
__global__ __launch_bounds__(32)
void k_gemm64(const float* __restrict__ A, const float* __restrict__ B,
              const float* __restrict__ bias, float* __restrict__ C,
              int K, int N, int act) {
  int nt = blockIdx.x, mt = blockIdx.y;
  int lane = threadIdx.x & 31;
  int l16 = lane & 15, hi = lane >> 4;
  int nb = nt * 16, mb = mt * 16;
  v8f c = {};
  const float* Ar = A + (size_t)(mb + l16) * K;
  const float* Bp = B + nb + l16;
  int Kfull = K & ~3;
  int k = 0;
  for (; k < Kfull; k += 4) {
    int kk = k + 2 * hi;
    v2f a = *(const v2f*)(Ar + kk);
    v2f b;
    b.x = Bp[(size_t)kk * N];
    b.y = Bp[(size_t)(kk + 1) * N];
    c = WMMA_F32(a, b, c);
  }
  if (k < K) {                       // K even; kk and kk+1 valid together
    int kk = k + 2 * hi;
    bool ok = kk < K;
    int kkc = ok ? kk : (K - 2);
    v2f a = *(const v2f*)(Ar + kkc);
    v2f b;
    b.x = Bp[(size_t)kkc * N];
    b.y = Bp[(size_t)(kkc + 1) * N];
    if (!ok) { a.x = 0.f; a.y = 0.f; b.x = 0.f; b.y = 0.f; }
    c = WMMA_F32(a, b, c);
  }
  float bv = bias[nb + l16];
#pragma unroll
  for (int r = 0; r < 8; ++r) {
    int m = mb + r + 8 * hi;
    float v = c[r] + bv;
    if (act == 1) v = (v > 0.f) ? v : expm1f(v);
    C[(size_t)m * N + nb + l16] = v;
  }
}

// ---------------------------------------------------------------------------
// 4. Expert logits + softmax -> w:(64,4)
// ---------------------------------------------------------------------------
__global__ void k_gate_softmax(const float* __restrict__ h2, const float* __restrict__ W,
                               const float* __restrict__ b, float* __restrict__ out) {
  int n = blockIdx.x * blockDim.x + threadIdx.x;
  if (n >= BS) return;
  float l[EE];
  for (int e = 0; e < EE; ++e) l[e] = b[e];
  const float* hr = h2 + (size_t)n * HG;
  for (int k = 0; k < HG; ++k) {
    float hv = hr[k];
#pragma unroll
    for (int e = 0; e < EE; ++e) l[e] += hv * W[k * EE + e];
  }
  float mx = fmaxf(fmaxf(l[0], l[1]), fmaxf(l[2], l[3]));
  float s = 0.f;
  for (int e = 0; e < EE; ++e) { l[e] = expf(l[e] - mx); s += l[e]; }
  for (int e = 0; e < EE; ++e) out[n * EE + e] = l[e] / s;
}

// ---------------------------------------------------------------------------
// 5. AS[n,t,v,w] = sum_e wgt[n,e] * A[e,t,v,w]
// ---------------------------------------------------------------------------
__global__ void k_build_AS(const float* __restrict__ wgt, const float* __restrict__ A,
                           float* __restrict__ AS) {
  int i = blockIdx.x * blockDim.x + threadIdx.x;
  if (i >= BS * TT * VV * VV) return;
  int p = i % (TT * VV * VV);
  int n = i / (TT * VV * VV);
  float acc = 0.f;
#pragma unroll
  for (int e = 0; e < EE; ++e) acc += wgt[n * EE + e] * A[(size_t)e * TT * VV * VV + p];
  AS[i] = acc;
}

// ---------------------------------------------------------------------------
// 6. y[n,c,t,w] = sum_v x[n,c,t,v] * AS[n,t,v,w]   (AS tile cached in LDS)
// ---------------------------------------------------------------------------
__global__ void k_apply_AS(const float* __restrict__ in, const float* __restrict__ AS,
                           float* __restrict__ out, int cin) {
  __shared__ float sA[VV * VV];
  int nt = blockIdx.x;                 // n*60 + t
  int n = nt / TT, t = nt % TT;
  const float* asp = AS + (size_t)nt * VV * VV;
  for (int i = threadIdx.x; i < VV * VV; i += blockDim.x) sA[i] = asp[i];
  __syncthreads();
  int total = cin * VV;
  for (int j = threadIdx.x; j < total; j += blockDim.x) {
    int c = j / VV, wv = j % VV;
    const float* xr = in + (((size_t)n * cin + c) * TT + t) * VV;
    float acc = 0.f;
#pragma unroll
    for (int v = 0; v < VV; ++v) acc += xr[v] * sA[v * VV + wv];
    out[(((size_t)n * cin + c) * TT + t) * VV + wv] = acc;
  }
}

// ---------------------------------------------------------------------------
// 7. Fused epilogue: out = PReLU( BN_t(tw@y + tb) + BN_r(rw@x + rb) )
//    Per sample: M=cout, N=1500 positions, K=cin. Two WMMA streams share K loop.
//    Out-of-range rows/cols are clamped (WMMA rows/cols are independent and
//    the extras are simply never stored) -> branch-free loads.
// ---------------------------------------------------------------------------
__global__ __launch_bounds__(32)
void k_outconv(const float* __restrict__ y, const float* __restrict__ x,
               const float* __restrict__ tw, const float* __restrict__ tb,
               const float* __restrict__ tg, const float* __restrict__ tbe,
               const float* __restrict__ trm, const float* __restrict__ trv,
               const float* __restrict__ rw, const float* __restrict__ rb,
               const float* __restrict__ rg, const float* __restrict__ rbe,
               const float* __restrict__ rrm, const float* __restrict__ rrv,
               const float* __restrict__ pa, float* __restrict__ out,
               int cin, int cout) {
  int nt = blockIdx.x, mt = blockIdx.y, n = blockIdx.z;
  int lane = threadIdx.x & 31;
  int l16 = lane & 15, hi = lane >> 4;
  int nb = nt * 16, mb = mt * 16;
  int m = mb + l16;
  int mc = (m < cout) ? m : (cout - 1);          // clamp: rows independent
  int pos = nb + l16;
  int pc = (pos < TV) ? pos : (TV - 1);          // clamp: cols independent
  const float* twp = tw + (size_t)mc * cin;
  const float* rwp = rw + (size_t)mc * cin;
  const float* ynp = y + (size_t)n * cin * TV + pc;
  const float* xnp = x + (size_t)n * cin * TV + pc;
  v8f cm = {}, cr = {};
  int Kfull = cin & ~3;
  int k = 0;
  for (; k < Kfull; k += 4) {
    int kk = k + 2 * hi;
    v2f am = *(const v2f*)(twp + kk);
    v2f ar = *(const v2f*)(rwp + kk);
    v2f bm, br;
    bm.x = ynp[(size_t)kk * TV];
    bm.y = ynp[(size_t)(kk + 1) * TV];
    br.x = xnp[(size_t)kk * TV];
    br.y = xnp[(size_t)(kk + 1) * TV];
    cm = WMMA_F32(am, bm, cm);
    cr = WMMA_F32(ar, br, cr);
  }
  if (k < cin) {                                  // cin=6 tail; cin even
    int kk = k + 2 * hi;
    bool ok = kk < cin;
    int kkc = ok ? kk : (cin - 2);
    v2f am = *(const v2f*)(twp + kkc);
    v2f ar = *(const v2f*)(rwp + kkc);
    v2f bm, br;
    bm.x = ynp[(size_t)kkc * TV];
    bm.y = ynp[(size_t)(kkc + 1) * TV];
    br.x = xnp[(size_t)kkc * TV];
    br.y = xnp[(size_t)(kkc + 1) * TV];
    if (!ok) {
      am.x = 0.f; am.y = 0.f; ar.x = 0.f; ar.y = 0.f;
      bm.x = 0.f; bm.y = 0.f; br.x = 0.f; br.y = 0.f;
    }
    cm = WMMA_F32(am, bm, cm);
    cr = WMMA_F32(ar, br, cr);
  }
  if (pos >= TV) return;
  float alpha = pa[0];
  float* on = out + (size_t)n * cout * TV;
#pragma unroll
  for (int r = 0; r < 8; ++r) {
    int o = mb + r + 8 * hi;
    if (o < cout) {
      float invt = tg[o] * rsqrtf(trv[o] + 1e-5f);
      float vm = (cm[r] + tb[o]) * invt + (tbe[o] - trm[o] * invt);
      float invr = rg[o] * rsqrtf(rrv[o] + 1e-5f);
      float vr = (cr[r] + rb[o]) * invr + (rbe[o] - rrm[o] * invr);
      float s = vm + vr;
      on[(size_t)o * TV + pos] = (s >= 0.f) ? s : alpha * s;
    }
  }
}

// ---------------------------------------------------------------------------
// 8. (64,6,60,25) -> (64,60,25,6) for the TCN stage
// ---------------------------------------------------------------------------
__global__ void k_transpose(const float* __restrict__ g, float* __restrict__ o) {
  int i = blockIdx.x * blockDim.x + threadIdx.x;
  if (i >= BS * TT * VV * FD) return;
  int f = i % FD;
  int v = (i / FD) % VV;
  int t = (i / (FD * VV)) % TT;
  int n = i / (FD * VV * TT);
  o[i] = g[(((size_t)n * FD + f) * TT + t) * VV + v];
}

// ---------------------------------------------------------------------------
// 9. 3x3 SAME conv (NCHW, H=25, W=6) + BN + exact GELU
// ---------------------------------------------------------------------------
__global__ void k_tcn(const float* __restrict__ x, const float* __restrict__ w,
                      const float* __restrict__ b, const float* __restrict__ g,
                      const float* __restrict__ be, const float* __restrict__ rm,
                      const float* __restrict__ rv, float* __restrict__ out,
                      int ci, int co) {
  int i = blockIdx.x * blockDim.x + threadIdx.x;
  int total = BS * co * VV * FD;
  if (i >= total) return;
  int wq = i % FD;
  int hq = (i / FD) % VV;
  int o  = (i / (FD * VV)) % co;
  int n  = i / (FD * VV * co);
  float acc = b[o];
  for (int c = 0; c < ci; ++c) {
    const float* xp = x + ((size_t)n * ci + c) * VV * FD;
    const float* wp = w + ((size_t)o * ci + c) * 9;
#pragma unroll
    for (int kh = 0; kh < 3; ++kh) {
      int hh = hq + kh - 1;
      if (hh < 0 || hh >= VV) continue;
#pragma unroll
      for (int kw = 0; kw < 3; ++kw) {
        int ww = wq + kw - 1;
        if (ww < 0 || ww >= FD) continue;
        acc += xp[hh * FD + ww] * wp[kh * 3 + kw];
      }
    }
  }
  float inv = g[o] * rsqrtf(rv[o] + 1e-5f);
  float v = acc * inv + (be[o] - rm[o] * inv);
  out[i] = 0.5f * v * (1.f + erff(v * 0.70710678118654752440f));
}

// ---------------------------------------------------------------------------
// Host: launch sequence
// ---------------------------------------------------------------------------
extern "C" void kernel_launch(void* const* d_in, const int* in_sizes, int n_in,
                              void* d_out, int out_size, void* d_ws, size_t ws_size,
                              hipStream_t stream) {
  (void)in_sizes; (void)n_in; (void)out_size; (void)ws_size;
  const int cin_arr[4]  = {6, 64, 256, 64};
  const int cout_arr[4] = {64, 256, 64, 6};

  auto F = [&](int i) -> const float* { return (const float*)d_in[i]; };

  // workspace layout (float offsets)
  float* W = (float*)d_ws;
  const size_t BIG = (size_t)BS * 256 * TV;          // 24,576,000
  float* BUF0 = W;
  float* BUF1 = W + BIG;
  float* BUF2 = W + 2 * BIG;                         // gcn y temp
  float* ASB  = W + 3 * BIG;                         // 2,400,000
  float* PART = ASB + (size_t)BS * TT * VV * VV;     // NSPLIT*64*256
  float* H1   = PART + (size_t)NSPLIT * BS * HG;
  float* H2   = H1 + BS * HG;
  float* WGT  = H2 + BS * HG;
  float* T0   = WGT + BS * EE;                       // (64,60,25,6)
  float* T1   = T0 + (size_t)BS * TT * VV * FD;      // (64,30,25,6)
  float* T2   = T1 + (size_t)BS * 30 * VV * FD;      // (64,15,25,6)
  float* T3   = T2 + (size_t)BS * 15 * VV * FD;      // (64,1,25,6) = (64,150)

  // 1. input assembly
  {
    int total = BS * FD * TT * VV;
    k_build_input<<<(total + 255) / 256, 256, 0, stream>>>(
        F(0), (const int*)d_in[1], F(4), BUF0);
  }

  // 2. four ST-GCN layers (ping-pong BUF0 <-> BUF1)
  float* cur = BUF0;
  float* nxt = BUF1;
  for (int L = 0; L < 4; ++L) {
    int base = 5 + 20 * L;
    int cin = cin_arr[L], cout = cout_arr[L];
    int K = cin * TV;
    int chunk = K / NSPLIT;          // exact, multiple of 8 for all layers

    // gating MLP
    k_gate0_splitk<<<dim3(16, NSPLIT), 32, 0, stream>>>(cur, F(base + 0), PART, K, chunk);
    k_reduce_elu<<<(BS * HG + 255) / 256, 256, 0, stream>>>(PART, F(base + 1), H1, NSPLIT);
    k_gemm64<<<dim3(16, 4), 32, 0, stream>>>(H1, F(base + 2), F(base + 3), H2, HG, HG, 1);
    k_gate_softmax<<<1, 64, 0, stream>>>(H2, F(base + 4), F(base + 5), WGT);

    // adjacency mix + graph contraction
    k_build_AS<<<(BS * TT * VV * VV + 255) / 256, 256, 0, stream>>>(WGT, F(base + 6), ASB);
    k_apply_AS<<<BS * TT, 256, 0, stream>>>(cur, ASB, BUF2, cin);

    // fused dual 1x1 conv + BN + residual + PReLU (WMMA)
    k_outconv<<<dim3((TV + 15) / 16, (cout + 15) / 16, BS), 32, 0, stream>>>(
        BUF2, cur,
        F(base + 7), F(base + 8), F(base + 9), F(base + 10), F(base + 11), F(base + 12),
        F(base + 13), F(base + 14), F(base + 15), F(base + 16), F(base + 17), F(base + 18),
        F(base + 19), nxt, cin, cout);

    float* tmp = cur; cur = nxt; nxt = tmp;
  }
  // after 4 layers, result (64,6,60,25) is in `cur` (== BUF0)

  // 3. TCN stage
  {
    int total = BS * TT * VV * FD;
    k_transpose<<<(total + 255) / 256, 256, 0, stream>>>(cur, T0);
  }
  {
    int t = 85;
    int tot0 = BS * 30 * VV * FD;
    k_tcn<<<(tot0 + 255) / 256, 256, 0, stream>>>(
        T0, F(t + 0), F(t + 1), F(t + 2), F(t + 3), F(t + 4), F(t + 5), T1, 60, 30);
    t = 91;
    int tot1 = BS * 15 * VV * FD;
    k_tcn<<<(tot1 + 255) / 256, 256, 0, stream>>>(
        T1, F(t + 0), F(t + 1), F(t + 2), F(t + 3), F(t + 4), F(t + 5), T2, 30, 15);
    t = 97;
    int tot2 = BS * 1 * VV * FD;
    k_tcn<<<(tot2 + 255) / 256, 256, 0, stream>>>(
        T2, F(t + 0), F(t + 1), F(t + 2), F(t + 3), F(t + 4), F(t + 5), T3, 15, 1);
  }

  // 4. heads: mu then logvar, concatenated in d_out
  {
    float* out = (float*)d_out;
    k_gemm64<<<dim3(16, 4), 32, 0, stream>>>(T3, F(103), F(104), out, VV * FD, 256, 0);
    k_gemm64<<<dim3(16, 4), 32, 0, stream>>>(T3, F(105), F(106), out + BS * 256, VV * FD, 256, 0);
  }
}